// GNN_21277267984701
// MI455X (gfx1250) — compile-verified
//
#include <hip/hip_runtime.h>
#include <math.h>

// ---------------------------------------------------------------------------
// GraphSAGE 3-layer GNN for MI455X (gfx1250, wave32).
// Scatter-bound workload; GEMMs use native FP32 WMMA (16x16x4) to keep the
// reference fp32 precision through three L2-normalize stages. Weight tiles
// are staged into LDS via global_load_async_to_lds_b128 (ASYNCcnt DMA) and
// read back with ds_load_b64, removing the 8x redundant B-side global loads.
// ---------------------------------------------------------------------------

#define N_NODES 50000
#define N_EDGES 500000
#define ROW_TILES (N_NODES / 16)   // 3125, exact
#define FC_IN ((long long)N_NODES * 64)
#define FC_BLOCKS 512
#define OUT_CH 5

typedef __attribute__((ext_vector_type(2))) float v2f;
typedef __attribute__((ext_vector_type(8))) float v8f;

// ---------------------------------------------------------------- utilities
__global__ void zero_f32(float* __restrict__ p, long long n) {
  long long i = (long long)blockIdx.x * blockDim.x + threadIdx.x;
  long long stride = (long long)gridDim.x * blockDim.x;
  for (; i < n; i += stride) p[i] = 0.0f;
}

// degree via float atomics: counts < 2^24 so sums are exact & order-independent
__global__ void deg_kernel(const int* __restrict__ ei, float* __restrict__ deg) {
  int e = blockIdx.x * blockDim.x + threadIdx.x;
  if (e < N_EDGES) atomicAdd(&deg[ei[N_EDGES + e]], 1.0f);
}

__global__ void invdeg_kernel(float* __restrict__ deg) {
  int i = blockIdx.x * blockDim.x + threadIdx.x;
  if (i < N_NODES) deg[i] = 1.0f / fmaxf(deg[i], 1.0f);
}

// one block per edge; coalesced row read, F atomic adds into dst row
template <int F>
__global__ void scatter_kernel(const int* __restrict__ ei,
                               const float* __restrict__ h,
                               float* __restrict__ agg) {
  int e = blockIdx.x;
  int c = threadIdx.x;
  int src = ei[e];
  int dst = ei[N_EDGES + e];
  atomicAdd(&agg[(size_t)dst * F + c], h[(size_t)src * F + c]);
}

// ------------------------------------------------------------- WMMA GEMM
// out[16r x 16c tile] = (agg*inv_deg) @ Wl^T + x @ Wr^T + bl
// One wave per 16x16 tile; K-loop of V_WMMA_F32_16X16X4_F32.
// A layout (ISA 7.12.2): lane l -> row M=l&15, K-half (l>>4)*2; comps = K,K+1.
// B layout (mirror):     lane l -> col N=l&15, same K-half split.
// C layout: comp r -> (M = r + (l<16?0:8), N = l&15).
// Weight tiles for this block's colTile are staged to LDS by async DMA.
template <int F_IN, int F_OUT>
__global__ void sage_gemm(const float* __restrict__ h_in,
                          const float* __restrict__ agg,
                          const float* __restrict__ inv_deg,
                          const float* __restrict__ Wl,   // [F_OUT][F_IN]
                          const float* __restrict__ bl,   // [F_OUT]
                          const float* __restrict__ Wr,   // [F_OUT][F_IN]
                          float* __restrict__ out) {      // [N][F_OUT]
  // +4 float pad: 4-dword bank stride -> 16 B-lanes hit distinct banks,
  // rows stay 16B-aligned for async B128 writes.
  __shared__ __align__(16) float sW[2][16][F_IN + 4];

  const int tid = threadIdx.x;
  const int wave = tid >> 5;
  const int lane = tid & 31;
  const int rowTile = blockIdx.x * 8 + wave;   // wave-uniform
  const int colTile = blockIdx.y;

  // ---- stage Wl/Wr tiles (16 x F_IN each) into LDS with async DMA.
  // chunks of 4 floats; TOTAL = 2*16*(F_IN/4) = 1024 or 2048 (multiple of 256)
  {
    constexpr int CH = F_IN / 4;
    constexpr int TOTAL = 2 * 16 * CH;
    const size_t gbase = (size_t)colTile * 16 * F_IN;
#pragma unroll 2
    for (int c = tid; c < TOTAL; c += 256) {
      const int mat = c / (16 * CH);
      const int rem = c - mat * (16 * CH);
      const int r = rem / CH;
      const int ch = rem - r * CH;
      const float* g = (mat == 0 ? Wl : Wr) + gbase + (size_t)r * F_IN + ch * 4;
      unsigned int laddr = (unsigned int)(size_t)&sW[mat][r][ch * 4];
      asm volatile("global_load_async_to_lds_b128 %0, %1, off"
                   :: "v"(laddr), "v"(g) : "memory");
    }
    asm volatile("s_wait_asynccnt 0x0" ::: "memory");
    __syncthreads();
  }

  if (rowTile < ROW_TILES) {                   // uniform per wave: EXEC all-1s
    const int m  = lane & 15;                  // A row within tile / B col
    const int kb = (lane >> 4) << 1;           // K-half: 0 or 2
    const int row = rowTile * 16 + m;          // node index for A

    const float dinv = inv_deg[row];
    const float* __restrict__ arow = agg  + (size_t)row * F_IN;
    const float* __restrict__ xrow = h_in + (size_t)row * F_IN;
    const float* sl = &sW[0][m][0];            // Wl tile row for col m (LDS)
    const float* sr = &sW[1][m][0];            // Wr tile row for col m (LDS)

    v8f acc = {};
    // mean @ Wl^T   (fold 1/deg into A)
    for (int k0 = 0; k0 < F_IN; k0 += 4) {
      v2f a = *(const v2f*)(arow + k0 + kb);
      a.x *= dinv; a.y *= dinv;
      v2f b = *(const v2f*)(sl + k0 + kb);     // ds_load_b64, conflict-free
      acc = __builtin_amdgcn_wmma_f32_16x16x4_f32(
          false, a, false, b, (short)0, acc, false, false);
    }
    // x @ Wr^T
    for (int k0 = 0; k0 < F_IN; k0 += 4) {
      v2f a = *(const v2f*)(xrow + k0 + kb);
      v2f b = *(const v2f*)(sr + k0 + kb);
      acc = __builtin_amdgcn_wmma_f32_16x16x4_f32(
          false, a, false, b, (short)0, acc, false, false);
    }

    // write back C tile + bias (row-wide normalize happens in a later pass)
    const int cn = colTile * 16 + (lane & 15);
    const int mbase = rowTile * 16 + ((lane >> 4) << 3);  // +0 or +8
    const float bias = bl[cn];
#pragma unroll
    for (int r = 0; r < 8; ++r)
      out[(size_t)(mbase + r) * F_OUT + cn] = acc[r] + bias;
  }
}

// --------------------------------------------- row L2-normalize + ReLU
// wave32 per node; normalize then relu (== relu then scale, scale > 0)
template <int F>
__global__ void norm_relu_kernel(float* __restrict__ h) {
  const int wave = threadIdx.x >> 5;
  const int lane = threadIdx.x & 31;
  const int node = blockIdx.x * (blockDim.x >> 5) + wave;
  if (node >= N_NODES) return;
  float* __restrict__ row = h + (size_t)node * F;
  float s = 0.0f;
  for (int c = lane; c < F; c += 32) { float v = row[c]; s += v * v; }
#pragma unroll
  for (int off = 16; off > 0; off >>= 1) s += __shfl_xor(s, off, 32);
  const float scale = 1.0f / fmaxf(sqrtf(s), 1e-12f);
  for (int c = lane; c < F; c += 32) row[c] = fmaxf(row[c] * scale, 0.0f);
}

// --------------------------------------------------- FC (5 x 3.2M) + softmax
// Deterministic: per-block partials via LDS tree reduce, then ordered sum.
__global__ void fc_partial_kernel(const float* __restrict__ fcW,
                                  const float* __restrict__ h3,
                                  float* __restrict__ partial) {
  __shared__ float sred[256];
  float acc[OUT_CH] = {0.f, 0.f, 0.f, 0.f, 0.f};
  long long i = (long long)blockIdx.x * blockDim.x + threadIdx.x;
  const long long stride = (long long)gridDim.x * blockDim.x;
  for (; i < FC_IN; i += stride) {
    const float v = h3[i];
    __builtin_prefetch(fcW + i + stride, 0, 1);  // global_prefetch_b8
#pragma unroll
    for (int o = 0; o < OUT_CH; ++o) acc[o] += fcW[(long long)o * FC_IN + i] * v;
  }
#pragma unroll
  for (int o = 0; o < OUT_CH; ++o) {
    sred[threadIdx.x] = acc[o];
    __syncthreads();
    for (int s2 = 128; s2 > 0; s2 >>= 1) {
      if (threadIdx.x < s2) sred[threadIdx.x] += sred[threadIdx.x + s2];
      __syncthreads();
    }
    if (threadIdx.x == 0) partial[(long long)blockIdx.x * OUT_CH + o] = sred[0];
    __syncthreads();
  }
}

__global__ void fc_final_kernel(const float* __restrict__ partial,
                                const float* __restrict__ fcb,
                                float* __restrict__ out) {
  if (threadIdx.x != 0) return;
  float logits[OUT_CH];
  for (int o = 0; o < OUT_CH; ++o) {
    float s = fcb[o];
    for (int b = 0; b < FC_BLOCKS; ++b) s += partial[(long long)b * OUT_CH + o];
    logits[o] = s;
  }
  float m = logits[0];
  for (int o = 1; o < OUT_CH; ++o) m = fmaxf(m, logits[o]);
  float z = 0.0f;
  for (int o = 0; o < OUT_CH; ++o) { logits[o] = expf(logits[o] - m); z += logits[o]; }
  for (int o = 0; o < OUT_CH; ++o) out[o] = logits[o] / z;
}

// ---------------------------------------------------------------------------
extern "C" void kernel_launch(void* const* d_in, const int* in_sizes, int n_in,
                              void* d_out, int out_size, void* d_ws, size_t ws_size,
                              hipStream_t stream) {
  const float* x   = (const float*)d_in[0];
  const int*   ei  = (const int*)d_in[1];   // [2][N_EDGES]: src row then dst row
  const float* W1l = (const float*)d_in[2];
  const float* b1  = (const float*)d_in[3];
  const float* W1r = (const float*)d_in[4];
  const float* W2l = (const float*)d_in[5];
  const float* b2  = (const float*)d_in[6];
  const float* W2r = (const float*)d_in[7];
  const float* W3l = (const float*)d_in[8];
  const float* b3  = (const float*)d_in[9];
  const float* W3r = (const float*)d_in[10];
  const float* fcW = (const float*)d_in[11];
  const float* fcb = (const float*)d_in[12];
  float* out = (float*)d_out;

  // workspace layout (floats)
  float* ws  = (float*)d_ws;
  float* deg = ws;                                    // N (padded)
  float* agg = deg + 50048;                           // N*256 max
  float* h1  = agg + (size_t)N_NODES * 256;           // N*256
  float* h2  = h1  + (size_t)N_NODES * 256;           // N*128
  float* h3  = h2  + (size_t)N_NODES * 128;           // N*64
  float* partial = h3 + (size_t)N_NODES * 64;         // FC_BLOCKS*5

  const dim3 blk256(256);

  // degrees (shared by all layers)
  zero_f32<<<256, blk256, 0, stream>>>(deg, N_NODES);
  deg_kernel<<<(N_EDGES + 255) / 256, blk256, 0, stream>>>(ei, deg);
  invdeg_kernel<<<(N_NODES + 255) / 256, blk256, 0, stream>>>(deg);

  const int gemmBlocksX = (ROW_TILES + 7) / 8;        // 391

  // ---- layer 1: 128 -> 256
  zero_f32<<<2048, blk256, 0, stream>>>(agg, (long long)N_NODES * 128);
  scatter_kernel<128><<<N_EDGES, 128, 0, stream>>>(ei, x, agg);
  sage_gemm<128, 256><<<dim3(gemmBlocksX, 16), blk256, 0, stream>>>(
      x, agg, deg, W1l, b1, W1r, h1);
  norm_relu_kernel<256><<<N_NODES / 8, blk256, 0, stream>>>(h1);

  // ---- layer 2: 256 -> 128
  zero_f32<<<2048, blk256, 0, stream>>>(agg, (long long)N_NODES * 256);
  scatter_kernel<256><<<N_EDGES, 256, 0, stream>>>(ei, h1, agg);
  sage_gemm<256, 128><<<dim3(gemmBlocksX, 8), blk256, 0, stream>>>(
      h1, agg, deg, W2l, b2, W2r, h2);
  norm_relu_kernel<128><<<N_NODES / 8, blk256, 0, stream>>>(h2);

  // ---- layer 3: 128 -> 64
  zero_f32<<<2048, blk256, 0, stream>>>(agg, (long long)N_NODES * 128);
  scatter_kernel<128><<<N_EDGES, 128, 0, stream>>>(ei, h2, agg);
  sage_gemm<128, 64><<<dim3(gemmBlocksX, 4), blk256, 0, stream>>>(
      h2, agg, deg, W3l, b3, W3r, h3);
  norm_relu_kernel<64><<<N_NODES / 8, blk256, 0, stream>>>(h3);

  // ---- FC + softmax
  fc_partial_kernel<<<FC_BLOCKS, blk256, 0, stream>>>(fcW, h3, partial);
  fc_final_kernel<<<1, 32, 0, stream>>>(partial, fcb, out);
}